// MCPBRNN_Generic_PETconstraint_Two_VariantOutputGate_10196252360850
// MI455X (gfx1250) — compile-verified
//
#include <hip/hip_runtime.h>
#include <cmath>

// ---------------------------------------------------------------------------
// MCPBRNN (H=1): sequential scalar recurrence over B*S=100k steps + parallel
// pre/post passes.  No matrix work exists (all weights are 1x1 scalars), so
// WMMA is not applicable; the CDNA5-specific path used here is the async
// global->LDS copy engine (ASYNCcnt) to stage the sequential input stream.
// ---------------------------------------------------------------------------

#define SPIN  365
#define TRAIN 5000
#define ML_C  2.9086f
#define SL_C  1.898f
#define LOG2E 1.4426950408889634f
#define CHUNK 64   // steps staged per LDS buffer (2 buffers, double-buffered)

__device__ __forceinline__ float fexp(float x) {
  return __builtin_amdgcn_exp2f(x * LOG2E);            // v_exp_f32
}
__device__ __forceinline__ float frcp(float x) {
  return __builtin_amdgcn_rcpf(x);                     // v_rcp_f32
}
__device__ __forceinline__ float fsig(float a) {       // sigmoid: 1 trans + add + 1 trans
  return frcp(1.0f + __builtin_amdgcn_exp2f(-a * LOG2E));
}

// --------------------------- kernel 1: pack / zero -------------------------
// Writes: bp_n, g_ib (all zero), g_ol (input-only sigmoid, hoisted out of the
// sequential loop), and zeros for rows b < time_lag of scan-written outputs.
__global__ void k_pack(const float* __restrict__ x, const int* __restrict__ tlp,
                       const float* __restrict__ r_yom, const float* __restrict__ r_gw,
                       const float* __restrict__ r_ylm, const float* __restrict__ r_yfm,
                       const float* __restrict__ b0_ylm, const float* __restrict__ w2_ylm,
                       float* __restrict__ out, int B)
{
  long b = (long)blockIdx.x * blockDim.x + threadIdx.x;
  if (b >= B) return;
  const int tl = tlp[0];

  const float e0 = fexp(r_yom[0]), e1 = fexp(r_gw[0]);
  const float e2 = fexp(r_ylm[0]), e3 = fexp(r_yfm[0]);
  const float ol1 = e2 * frcp(e0 + e1 + e2 + e3);

  out[4L * B + b] = 0.0f;   // bp_n  == 0 always
  out[6L * B + b] = 0.0f;   // g_ib  == 0 always

  float golv = 0.0f;
  if (b >= tl) {
    const float u2 = x[2 * b + 1];
    golv = ol1 * fsig(fmaf((u2 - ML_C) * (1.0f / SL_C), w2_ylm[0], b0_ylm[0]));
  } else {
    // zero-pad the skipped time_lag rows of the scan-written outputs
    out[0L * B + b] = 0.0f;           // h_n
    out[1L * B + b] = 0.0f;           // c_n
    out[2L * B + b] = 0.0f;           // l_n
    out[3L * B + b] = 0.0f;           // lc_n
    out[5L * B + b] = 0.0f;           // gw_n
    out[7L * B + b] = 0.0f;           // g_oo
    out[9L * B + b] = 0.0f;           // g_olc
    out[10L * B + b] = 0.0f;          // g_f
    out[11L * B + b] = 0.0f;          // g_oogw
    out[12L * B + 2 * b] = 0.0f;      // h_nout[:,0]
  }
  out[8L * B + b] = golv;             // g_ol (also consumed by the scan)
}

// --------------------------- kernel 2: obs std -----------------------------
__global__ void k_std(const float* __restrict__ y, int n0, int n1,
                      float* __restrict__ ws)
{
  __shared__ double ssum[256];
  __shared__ double ssq[256];
  double s = 0.0, q = 0.0;
  for (int i = n0 + threadIdx.x; i < n1; i += 256) {
    const double v = (double)y[i];
    s += v; q += v * v;
  }
  ssum[threadIdx.x] = s; ssq[threadIdx.x] = q;
  __syncthreads();
  for (int o = 128; o > 0; o >>= 1) {
    if (threadIdx.x < o) {
      ssum[threadIdx.x] += ssum[threadIdx.x + o];
      ssq[threadIdx.x]  += ssq[threadIdx.x + o];
    }
    __syncthreads();
  }
  if (threadIdx.x == 0) {
    const double n = (double)(n1 - n0);
    double stdv = 0.0;
    if (n > 1.5) {
      double var = (ssq[0] - ssum[0] * ssum[0] / n) / (n - 1.0);
      if (var < 0.0) var = 0.0;
      stdv = sqrt(var);
    }
    ws[0] = (float)stdv;
  }
}

// --------------------------- kernel 3: std fills ---------------------------
__global__ void k_fill(const int* __restrict__ tlp, const float* __restrict__ ws,
                       float* __restrict__ out, int B)
{
  long b = (long)blockIdx.x * blockDim.x + threadIdx.x;
  if (b >= B) return;
  const float v = (b >= tlp[0]) ? ws[0] : 0.0f;
  out[14L * B + b] = v;               // obs_std
  out[12L * B + 2 * b + 1] = v;       // h_nout[:,1]
}

// --------------------------- kernel 4: sequential scan ---------------------
#if __has_builtin(__builtin_amdgcn_global_load_async_to_lds_b64) && \
    __has_builtin(__builtin_amdgcn_global_load_async_to_lds_b32) && \
    __has_builtin(__builtin_amdgcn_s_wait_asynccnt)
#define HAVE_ASYNC 1
#else
#define HAVE_ASYNC 0
#endif

#if HAVE_ASYNC
// Exact builtin pointee types (from clang diagnostics):
//   b64: (v2i AS1*, v2i AS3*, imm, imm)   b32: (int AS1*, int AS3*, imm, imm)
typedef __attribute__((ext_vector_type(2))) int v2i_t;
#define GAS __attribute__((address_space(1)))
#define LAS __attribute__((address_space(3)))
// generic->AS1 / AS3 via integer (generic->LDS is a low-32-bit trunc on AMDGCN)
#define GP64(p) ((GAS v2i_t*)(unsigned long long)(p))
#define LP64(p) ((LAS v2i_t*)(unsigned int)(unsigned long long)(p))
#define GP32(p) ((GAS int*)(unsigned long long)(p))
#define LP32(p) ((LAS int*)(unsigned int)(unsigned long long)(p))
#endif

__global__ __launch_bounds__(32) void k_scan(
    const float* __restrict__ x, const int* __restrict__ tlp,
    const float* __restrict__ p_mean, const float* __restrict__ p_std,
    const float* __restrict__ r_yom, const float* __restrict__ r_gw,
    const float* __restrict__ r_ylm, const float* __restrict__ r_yfm,
    const float* __restrict__ b0_yom, const float* __restrict__ w1_yom,
    const float* __restrict__ b0_gw, const float* __restrict__ w1_gw,
    const float* __restrict__ wstd, float* __restrict__ out, int B)
{
  __shared__ float2 xb[2][CHUNK];   // (u1,u2) per step
  __shared__ float  ob[2][CHUNK];   // precomputed ol per step
  const int lane = (int)threadIdx.x;
  const int tl = tlp[0];
  const long N = (long)B - tl;
  if (N <= 0) return;

  const float* gol = out + 8L * B;  // g_ol written by k_pack (stream-serialized)

  const float pm  = p_mean[0];
  const float ips = frcp(p_std[0]);
  const float e0 = fexp(r_yom[0]), e1 = fexp(r_gw[0]);
  const float e2 = fexp(r_ylm[0]), e3 = fexp(r_yfm[0]);
  const float iden  = frcp(e0 + e1 + e2 + e3);
  const float oo1   = e0 * iden;
  const float oogw1 = e1 * iden;
  const float bb1 = b0_yom[0], ww1 = w1_yom[0];
  const float bb2 = b0_gw[0],  ww2 = w1_gw[0];
  const float stdv = wstd[0];

  float* O_H     = out + 0L * B;
  float* O_C     = out + 1L * B;
  float* O_L     = out + 2L * B;
  float* O_LC    = out + 3L * B;
  float* O_GW    = out + 5L * B;
  float* O_GOO   = out + 7L * B;
  float* O_GOLC  = out + 9L * B;
  float* O_GF    = out + 10L * B;
  float* O_GOOGW = out + 11L * B;
  float2* O_HN   = (float2*)(out + 12L * B);

  const long nch = (N + CHUNK - 1) / CHUNK;

  auto issue = [&](long ci, int buf) {
    const long base = ci * CHUNK;
    long i0 = base + lane;      if (i0 > N - 1) i0 = N - 1;
    long i1 = base + 32 + lane; if (i1 > N - 1) i1 = N - 1;
#if HAVE_ASYNC
    __builtin_amdgcn_global_load_async_to_lds_b64(GP64(x + 2 * (i0 + tl)), LP64(&xb[buf][lane]),      0, 0);
    __builtin_amdgcn_global_load_async_to_lds_b64(GP64(x + 2 * (i1 + tl)), LP64(&xb[buf][32 + lane]), 0, 0);
    __builtin_amdgcn_global_load_async_to_lds_b32(GP32(gol + (i0 + tl)),   LP32(&ob[buf][lane]),      0, 0);
    __builtin_amdgcn_global_load_async_to_lds_b32(GP32(gol + (i1 + tl)),   LP32(&ob[buf][32 + lane]), 0, 0);
#else
    const float2* x2 = (const float2*)x;
    xb[buf][lane]      = x2[i0 + tl];
    xb[buf][32 + lane] = x2[i1 + tl];
    ob[buf][lane]      = gol[i0 + tl];
    ob[buf][32 + lane] = gol[i1 + tl];
#endif
  };

  float c0 = 0.0f;
  issue(0, 0);
  for (long ci = 0; ci < nch; ++ci) {
    const int buf = (int)(ci & 1);
    if (ci + 1 < nch) {
      issue(ci + 1, buf ^ 1);
#if HAVE_ASYNC
      __builtin_amdgcn_s_wait_asynccnt(4);  // only chunk ci+1's 4 copies may remain
#endif
    } else {
#if HAVE_ASYNC
      __builtin_amdgcn_s_wait_asynccnt(0);
#endif
    }
#if !HAVE_ASYNC
    __syncthreads();                        // make lane stores visible to lane 0
#endif
    asm volatile("" ::: "memory");          // keep LDS reads below the wait

    if (lane == 0) {
      const long base = ci * CHUNK;
#pragma unroll
      for (int blk = 0; blk < CHUNK / 8; ++blk) {
        float2 xv[8]; float olv[8];
#pragma unroll
        for (int k = 0; k < 8; ++k) {       // batch LDS reads off the chain
          xv[k]  = xb[buf][blk * 8 + k];
          olv[k] = ob[buf][blk * 8 + k];
        }
#pragma unroll
        for (int k = 0; k < 8; ++k) {
          const long t = base + blk * 8 + k;
          if (t < N) {
            const float u1 = xv[k].x, u2 = xv[k].y, ol = olv[k];
            const float z    = (c0 - pm) * ips;
            const float oo   = oo1   * fsig(fmaf(z, ww1, bb1));
            const float oogw = oogw1 * fsig(fmaf(z, ww2, bb2));
            const float q    = u2 * frcp(c0);
            // ol - relu(ol - u2/c0) == min(ol, u2/c0) when c0 > 0
            const float olc  = (c0 > 0.0f) ? fminf(ol, q) : ol;
            const float f    = 1.0f - oo - oogw - olc;
            const long  b    = t + tl;
            const float h    = oo * c0;
            O_H[b]     = h;
            O_C[b]     = c0;
            O_L[b]     = ol * c0;
            O_LC[b]    = olc * c0;
            O_GW[b]    = oogw * c0;
            O_GOO[b]   = oo;
            O_GOLC[b]  = olc;
            O_GF[b]    = f;
            O_GOOGW[b] = oogw;
            O_HN[b]    = make_float2(h, stdv);
            c0 = fmaf(f, c0, u1);           // ib = 0 => full inflow
          }
        }
      }
    }
  }
}

// --------------------------------- launch ----------------------------------
extern "C" void kernel_launch(void* const* d_in, const int* in_sizes, int n_in,
                              void* d_out, int out_size, void* d_ws, size_t ws_size,
                              hipStream_t stream)
{
  (void)n_in; (void)out_size; (void)ws_size;
  const float* x      = (const float*)d_in[0];
  const float* y_obs  = (const float*)d_in[1];
  const float* p_mean = (const float*)d_in[2];
  const float* p_std  = (const float*)d_in[3];
  const float* r_yom  = (const float*)d_in[4];
  const float* r_gw   = (const float*)d_in[5];
  const float* r_ylm  = (const float*)d_in[6];
  const float* r_yfm  = (const float*)d_in[7];
  const float* b0_yom = (const float*)d_in[8];
  const float* w1_yom = (const float*)d_in[9];
  const float* b0_gw  = (const float*)d_in[10];
  const float* w1_gw  = (const float*)d_in[11];
  const float* b0_ylm = (const float*)d_in[12];
  const float* w2_ylm = (const float*)d_in[13];
  /* d_in[14] = epoch (unused) */
  const int*   tl     = (const int*)d_in[15];

  float* out = (float*)d_out;
  float* ws  = (float*)d_ws;          // ws[0] = obs std scalar

  const int B = in_sizes[0] / 2;      // x is [B, 1, 2]
  const int nblk = (B + 255) / 256;

  k_pack<<<nblk, 256, 0, stream>>>(x, tl, r_yom, r_gw, r_ylm, r_yfm,
                                   b0_ylm, w2_ylm, out, B);

  int n1 = in_sizes[1] < TRAIN ? in_sizes[1] : TRAIN;
  k_std<<<1, 256, 0, stream>>>(y_obs, SPIN, n1, ws);

  k_fill<<<nblk, 256, 0, stream>>>(tl, ws, out, B);

  k_scan<<<1, 32, 0, stream>>>(x, tl, p_mean, p_std, r_yom, r_gw, r_ylm, r_yfm,
                               b0_yom, w1_yom, b0_gw, w1_gw, ws, out, B);
}